// TimeAwareScaledDotProductAttention_22823456211555
// MI455X (gfx1250) — compile-verified
//
#include <hip/hip_runtime.h>

typedef __attribute__((ext_vector_type(2))) float v2f;
typedef __attribute__((ext_vector_type(8))) float v8f;

// Problem constants from the reference
#define BB 8
#define HH 4
#define LL 256
#define DD 32
#define NEG_BIG (-4294967295.0f)  // -2^32 + 1

__global__ __launch_bounds__(256)
void ta_attn_kernel(const float* __restrict__ Q, const float* __restrict__ K,
                    const float* __restrict__ V, const float* __restrict__ tmK,
                    const float* __restrict__ tmV, const float* __restrict__ am,
                    const unsigned char* __restrict__ pm,
                    float* __restrict__ out_x, float* __restrict__ out_attn)
{
    const int qt  = blockIdx.x;      // 0..15  (q tile of 16 rows)
    const int bh  = blockIdx.y;      // 0..31  (b*H + h)
    const int b   = bh >> 2;         // H == 4
    const int q0  = qt * 16;
    const int tid = threadIdx.x;
    const int lane = tid & 31;
    const int wv   = tid >> 5;       // wave id 0..7

    __shared__ float sQ[16][DD];         // 2 KB  Q tile
    __shared__ float sE[16][LL + 1];     // ~16 KB energy / attn tile (padded)
    __shared__ float sRed[16][17];       // softmax reduction scratch
    __shared__ float sX[16][DD];         // partial x accumulation

    const float* Qb = Q + (size_t)(bh * LL + q0) * DD;
    const float* Kb = K + (size_t)bh * LL * DD;
    const float* Vb = V + (size_t)bh * LL * DD;
    const size_t tmBase = (size_t)(bh * LL + q0) * LL * DD;  // + q*L*D + k*D

    // ---- Phase 1: load Q tile to LDS ----
    for (int i = tid; i < 16 * DD; i += 256)
        sQ[i >> 5][i & 31] = Qb[i];
    __syncthreads();

    // ---- Phase 2: E = Q * K^T via V_WMMA_F32_16X16X4_F32 ----
    {
        const int m  = lane & 15;    // row (A) / col (B) index
        const int hi = lane >> 4;    // selects K-pair within 4-wide step
        v2f a[8];
#pragma unroll
        for (int dc = 0; dc < 8; ++dc) {
            const int d0 = dc * 4 + hi * 2;
            a[dc].x = sQ[m][d0];
            a[dc].y = sQ[m][d0 + 1];
        }
#pragma unroll
        for (int t = 0; t < 2; ++t) {
            const int k0 = (wv * 2 + t) * 16;           // this wave's N tile
            const float* kr = Kb + (size_t)(k0 + m) * DD;
            v8f c = {0.f, 0.f, 0.f, 0.f, 0.f, 0.f, 0.f, 0.f};
#pragma unroll
            for (int dc = 0; dc < 8; ++dc) {
                const int d0 = dc * 4 + hi * 2;
                v2f bf;
                bf.x = kr[d0];
                bf.y = kr[d0 + 1];
                c = __builtin_amdgcn_wmma_f32_16x16x4_f32(
                        false, a[dc], false, bf, (short)0, c, false, false);
            }
#pragma unroll
            for (int r = 0; r < 8; ++r)
                sE[r + hi * 8][k0 + m] = c[r];
        }
    }
    __syncthreads();

    // ---- Phase 3: energy += tmK[q,k,:] . Q[q,:]; scale + masks ----
    {
        const int k = tid;                               // 0..255
        const float inv_scale = 0.17677669529663687f;    // 1/sqrt(32)
        const bool pad = pm[b * LL + k] != 0;
        const float* tp = tmK + tmBase + (size_t)k * DD;
#pragma unroll 1
        for (int q = 0; q < 16; ++q) {
            const float4* t4 = (const float4*)(tp + (size_t)q * LL * DD);
            if (q < 15)
                __builtin_prefetch(tp + (size_t)(q + 1) * LL * DD, 0, 1);
            float e = 0.f;
#pragma unroll
            for (int c4 = 0; c4 < 8; ++c4) {
                const float4 tv = t4[c4];
                const float4 qv = *(const float4*)(&sQ[q][c4 * 4]);
                e += tv.x * qv.x + tv.y * qv.y + tv.z * qv.z + tv.w * qv.w;
            }
            const float eng = (sE[q][k] + e) * inv_scale + am[(q0 + q) * LL + k];
            sE[q][k] = pad ? NEG_BIG : eng;
        }
    }
    __syncthreads();

    // ---- Phase 4: row softmax over k ----
    {
        const int r = tid >> 4;      // row 0..15
        const int c = tid & 15;      // 16 threads per row, 16 cols each
        float mx = -3.4e38f;
#pragma unroll
        for (int j = 0; j < 16; ++j)
            mx = fmaxf(mx, sE[r][c * 16 + j]);
        sRed[r][c] = mx;
        __syncthreads();
        if (c == 0) {
            float mm = sRed[r][0];
#pragma unroll
            for (int j = 1; j < 16; ++j) mm = fmaxf(mm, sRed[r][j]);
            sRed[r][16] = mm;
        }
        __syncthreads();
        const float rm = sRed[r][16];
        float s = 0.f;
#pragma unroll
        for (int j = 0; j < 16; ++j) {
            const float e = __expf(sE[r][c * 16 + j] - rm);
            sE[r][c * 16 + j] = e;
            s += e;
        }
        sRed[r][c] = s;
        __syncthreads();
        if (c == 0) {
            float ss = 0.f;
#pragma unroll
            for (int j = 0; j < 16; ++j) ss += sRed[r][j];
            sRed[r][16] = 1.f / ss;
        }
        __syncthreads();
        const float rinv = sRed[r][16];
#pragma unroll
        for (int j = 0; j < 16; ++j)
            sE[r][c * 16 + j] *= rinv;
    }
    __syncthreads();

    // ---- write attn output (coalesced) ----
    for (int i = tid; i < 16 * LL; i += 256) {
        const int q = i >> 8, k = i & 255;
        out_attn[(size_t)(bh * LL + q0 + q) * LL + k] = sE[q][k];
    }

    // ---- Phase 5: x[q,:] = sum_k attn[q,k] * (V[k,:] + tmV[q,k,:]) ----
    {
        const int dc = tid & 7;            // float4 chunk of D
        const int q  = (tid >> 3) & 15;    // q row
        const int kh = tid >> 7;           // k half 0/1
        const float* tvb = tmV + tmBase + (size_t)q * LL * DD + dc * 4;
        const float* vvb = Vb + dc * 4;
        float ax = 0.f, ay = 0.f, az = 0.f, aw = 0.f;
        const int kend = kh * 128 + 128;
#pragma unroll 4
        for (int k = kh * 128; k < kend; ++k) {
            const float a = sE[q][k];
            const float4 tv = *(const float4*)(tvb + (size_t)k * DD);
            const float4 vv = *(const float4*)(vvb + (size_t)k * DD);
            ax += a * (tv.x + vv.x);
            ay += a * (tv.y + vv.y);
            az += a * (tv.z + vv.z);
            aw += a * (tv.w + vv.w);
        }
        if (kh == 0) {
            sX[q][dc * 4 + 0] = ax;
            sX[q][dc * 4 + 1] = ay;
            sX[q][dc * 4 + 2] = az;
            sX[q][dc * 4 + 3] = aw;
        }
        __syncthreads();
        if (kh == 1) {
            float4 o;
            o.x = sX[q][dc * 4 + 0] + ax;
            o.y = sX[q][dc * 4 + 1] + ay;
            o.z = sX[q][dc * 4 + 2] + az;
            o.w = sX[q][dc * 4 + 3] + aw;
            *(float4*)(out_x + (size_t)(bh * LL + q0 + q) * DD + dc * 4) = o;
        }
    }
}

extern "C" void kernel_launch(void* const* d_in, const int* in_sizes, int n_in,
                              void* d_out, int out_size, void* d_ws, size_t ws_size,
                              hipStream_t stream) {
    const float* Q   = (const float*)d_in[0];
    const float* K   = (const float*)d_in[1];
    const float* V   = (const float*)d_in[2];
    const float* tmK = (const float*)d_in[3];
    const float* tmV = (const float*)d_in[4];
    const float* am  = (const float*)d_in[5];
    const unsigned char* pm = (const unsigned char*)d_in[6];

    float* out_x    = (float*)d_out;                    // [B,H,L,D]
    float* out_attn = out_x + (size_t)BB * HH * LL * DD; // [B,H,L,L]

    dim3 grid(LL / 16, BB * HH);
    ta_attn_kernel<<<grid, 256, 0, stream>>>(Q, K, V, tmK, tmV, am, pm,
                                             out_x, out_attn);
}